// textEmbedding_77232101917230
// MI455X (gfx1250) — compile-verified
//
#include <hip/hip_runtime.h>

// Problem constants from the reference: B=256, T=2048, V=32000, C=128.
//   out[b, c, t] = W[c, idx[b, t]] + bias[c]        (out is (B, C, T) fp32)
//
// Memory-bound embedding gather:
//   - 256 MB output stream (HBM floor ~11us @ 23.3 TB/s)  -> NT b128 stores
//   - W (16.4 MB) is L2-resident (192 MB L2)              -> temporal gather loads
//   - index tile staged via CDNA5 async global->LDS copy  (ASYNCcnt path)

#define B_  256
#define T_  2048
#define V_  32000
#define C_  128

#define THREADS     256      // 8 wave32s per block
#define T_PER_THR   4        // 4 consecutive t per thread -> b128 stores
#define BT_PER_BLK  (THREADS * T_PER_THR)   // 1024 (b,t) pairs per block
#define C_SPLIT     4        // c dimension split across blockIdx.y
#define C_PER_BLK   (C_ / C_SPLIT)          // 32 c-rows per block

typedef float  vfloat4 __attribute__((ext_vector_type(4)));
typedef int    vint4   __attribute__((ext_vector_type(4)));

__global__ __launch_bounds__(THREADS) void textEmbedding_gather_kernel(
    const int*   __restrict__ idx,    // (B, 1, T) int32
    const float* __restrict__ W,      // (C, V) fp32, row-major
    const float* __restrict__ bias,   // (C,) fp32
    float*       __restrict__ out)    // (B, C, T) fp32
{
    __shared__ int lds_idx[BT_PER_BLK];   // 4 KB index tile

    const int tid = threadIdx.x;
    const long long bt0 = (long long)blockIdx.x * BT_PER_BLK;

    // ---- CDNA5 async copy: 4 KB of indices, global -> LDS, 16 B per lane ----
    // (gfx1250 GLOBAL_LOAD_ASYNC_TO_LDS_B128, tracked by ASYNCcnt; each lane
    //  later reads back only its own 16 B, so a per-wave asynccnt wait is
    //  sufficient ordering — no workgroup barrier required.)
    unsigned lds_dst = (unsigned)(size_t)(void*)lds_idx + (unsigned)tid * 16u;
    const int* gsrc  = idx + bt0 + (long long)tid * T_PER_THR;
    asm volatile("global_load_async_to_lds_b128 %0, %1, off"
                 :: "v"(lds_dst), "v"(gsrc)
                 : "memory");
    asm volatile("s_wait_asynccnt 0" ::: "memory");

    // Read back this lane's 4 vocabulary indices from LDS (ds_load_b128).
    vint4 v4 = *reinterpret_cast<const vint4*>(&lds_idx[tid * T_PER_THR]);

    const long long bt = bt0 + (long long)tid * T_PER_THR;
    const int b_idx = (int)(bt >> 11);        // bt / T_   (T_ = 2048)
    const int t0    = (int)(bt & (T_ - 1));   // bt % T_   (multiple of 4 -> 16B aligned)

    const int c0 = blockIdx.y * C_PER_BLK;
    float* outBase = out + (size_t)b_idx * C_ * T_ + (size_t)c0 * T_ + t0;

    #pragma unroll 2
    for (int cc = 0; cc < C_PER_BLK; ++cc) {
        const int c = c0 + cc;
        const float* Wc = W + (size_t)c * V_;   // one 128 KB row, L2-resident
        const float  bc = bias[c];              // wave-uniform -> s_load

        vfloat4 r;
        r.x = Wc[v4.x] + bc;                    // scattered 4B gathers -> L2 hits
        r.y = Wc[v4.y] + bc;
        r.z = Wc[v4.z] + bc;
        r.w = Wc[v4.w] + bc;

        // Streaming output: non-temporal b128 store, coalesced across the wave,
        // keeps the 256 MB output from evicting W out of L2.
        __builtin_nontemporal_store(
            r, reinterpret_cast<vfloat4*>(outBase + (size_t)cc * T_));
    }
}

extern "C" void kernel_launch(void* const* d_in, const int* in_sizes, int n_in,
                              void* d_out, int out_size, void* d_ws, size_t ws_size,
                              hipStream_t stream) {
    const int*   idx  = (const int*)  d_in[0];   // inputs: (B,1,T) int32
    const float* W    = (const float*)d_in[1];   // W: (C,V) fp32
    const float* bias = (const float*)d_in[2];   // b: (C,) fp32
    float*       out  = (float*)      d_out;     // (B,C,T) fp32

    dim3 grid((B_ * T_) / BT_PER_BLK, C_SPLIT);  // (512, 4)
    dim3 block(THREADS);                         // 256 threads = 8 wave32s
    textEmbedding_gather_kernel<<<grid, block, 0, stream>>>(idx, W, bias, out);
}